// ShiftConv_26645977104667
// MI455X (gfx1250) — compile-verified
//
#include <hip/hip_runtime.h>

typedef __attribute__((ext_vector_type(2))) float v2f;
typedef __attribute__((ext_vector_type(8))) float v8f;

#define B_SZ   32
#define CIN    256
#define COUT   512
#define HH     64
#define WW     64
#define HWSZ   4096      // 64*64
#define NG     9
#define KPAD   12        // K=9 padded to 12 -> 3 x wmma 16x16x4
#define IN_CPG 28
#define OUT_CPG 56

// ---------------------------------------------------------------------------
// Pass 1: W_eff[o][g] = sum of w_pw[o, c] over columns c in shift-group g.
// Groups of 56 columns; group 4 (centre) also absorbs tail columns 504..511.
// Output padded to [512][12] with zeros so K-dim is a multiple of 4.
// ---------------------------------------------------------------------------
__global__ void weff_kernel(const float* __restrict__ w_pw,
                            float* __restrict__ weff) {
    int i = blockIdx.x * blockDim.x + threadIdx.x;
    if (i >= COUT * KPAD) return;
    int o = i / KPAD;
    int g = i - o * KPAD;
    float sum = 0.0f;
    if (g < NG) {
        const float* row = w_pw + (size_t)o * COUT;
        int c0 = g * OUT_CPG;
        #pragma unroll 8
        for (int c = 0; c < OUT_CPG; ++c) sum += row[c0 + c];
        if (g == 4) {
            #pragma unroll
            for (int c = NG * OUT_CPG; c < COUT; ++c) sum += row[c];  // 504..511
        }
    }
    weff[i] = sum;
}

// ---------------------------------------------------------------------------
// Pass 2: s[b,g,h,w] = sum over 28 input channels of group g.
// float4 (b128) loads, 4 pixels per thread; reads 132 MB of x.
// ---------------------------------------------------------------------------
__global__ void gsum_kernel(const float* __restrict__ x,
                            float* __restrict__ s) {
    int i = blockIdx.x * blockDim.x + threadIdx.x;   // < 32*9*1024
    int pix4 = i & 1023;          // 4096/4 quads per (b,g)
    int bg   = i >> 10;
    int b    = bg / NG;
    int g    = bg - b * NG;
    const float* xp = x + (size_t)(b * CIN + g * IN_CPG) * HWSZ + pix4 * 4;
    float4 a = make_float4(0.f, 0.f, 0.f, 0.f);
    #pragma unroll
    for (int k = 0; k < IN_CPG; ++k) {
        float4 v = *(const float4*)(xp + (size_t)k * HWSZ);
        a.x += v.x; a.y += v.y; a.z += v.z; a.w += v.w;
    }
    *(float4*)(s + (size_t)bg * HWSZ + pix4 * 4) = a;
}

// ---------------------------------------------------------------------------
// Pass 3: out[b,o,h,w] = sum_g W_eff[o,g] * s[b,g, h+dh(g)-1, w+dw(g)-1]
// One block (8 waves) per (b,h) image row: stage im2col tile [64 px][12 K]
// in LDS, then each wave computes 4 oc-tiles x 4 px-tiles of 16x16 output
// via 3 x v_wmma_f32_16x16x4_f32 each.
// ---------------------------------------------------------------------------
__global__ void __launch_bounds__(256)
shiftmm_kernel(const float* __restrict__ s,
               const float* __restrict__ weff,
               float* __restrict__ out) {
    __shared__ __align__(16) float lds[WW * KPAD];   // 64*12 floats = 3 KB

    int bh = blockIdx.x;
    int b  = bh >> 6;
    int h  = bh & 63;
    int tid = threadIdx.x;

    // Build B-matrix staging tile: lds[w*12 + k] = shifted s value (0 on pad)
    for (int idx = tid; idx < WW * KPAD; idx += 256) {
        int k = idx >> 6;          // k-major so global reads are row-coalesced
        int w = idx & 63;
        float val = 0.0f;
        if (k < NG) {
            int dh = k / 3;
            int hh = h + dh - 1;
            int ww = w + (k - dh * 3) - 1;
            if ((unsigned)hh < (unsigned)HH && (unsigned)ww < (unsigned)WW)
                val = s[((size_t)(b * NG + k) << 12) + hh * WW + ww];
        }
        lds[w * KPAD + k] = val;
    }
    __syncthreads();

    int wave = tid >> 5;
    int lane = tid & 31;
    int half = lane >> 4;     // f32 WMMA: lanes 16-31 hold K=2,3 of each step
    int ln   = lane & 15;

    // B operands: 4 pixel tiles x 3 K-steps, each a b64 LDS read per lane
    v2f Bv[4][3];
    #pragma unroll
    for (int t = 0; t < 4; ++t) {
        int p = t * 16 + ln;
        #pragma unroll
        for (int ks = 0; ks < 3; ++ks)
            Bv[t][ks] = *(const v2f*)&lds[p * KPAD + ks * 4 + 2 * half];
    }

    #pragma unroll
    for (int oct = 0; oct < 4; ++oct) {
        int otile = wave * 4 + oct;           // 0..31 oc-tiles of 16
        int orow  = otile * 16 + ln;          // M row = output channel
        v2f Av[3];
        #pragma unroll
        for (int ks = 0; ks < 3; ++ks)
            Av[ks] = *(const v2f*)&weff[orow * KPAD + ks * 4 + 2 * half];

        #pragma unroll
        for (int t = 0; t < 4; ++t) {
            v8f acc = {0.f, 0.f, 0.f, 0.f, 0.f, 0.f, 0.f, 0.f};
            #pragma unroll
            for (int ks = 0; ks < 3; ++ks)
                acc = __builtin_amdgcn_wmma_f32_16x16x4_f32(
                    /*neg_a=*/false, Av[ks],
                    /*neg_b=*/false, Bv[t][ks],
                    /*c_mod=*/(short)0, acc,
                    /*reuse_a=*/false, /*reuse_b=*/false);

            // D layout: VGPR r -> row M = r + 8*half, col N = ln
            float* ob = out + (size_t)(b * COUT + otile * 16) * HWSZ
                            + h * WW + t * 16;
            #pragma unroll
            for (int r = 0; r < 8; ++r)
                ob[(size_t)(r + 8 * half) * HWSZ + ln] = acc[r];
        }
    }
}

// ---------------------------------------------------------------------------
extern "C" void kernel_launch(void* const* d_in, const int* in_sizes, int n_in,
                              void* d_out, int out_size, void* d_ws, size_t ws_size,
                              hipStream_t stream) {
    (void)in_sizes; (void)n_in; (void)out_size; (void)ws_size;
    const float* x    = (const float*)d_in[0];   // (32,256,64,64) f32
    const float* w_pw = (const float*)d_in[1];   // (512,512) f32
    float* out = (float*)d_out;                  // (32,512,64,64) f32

    float* s    = (float*)d_ws;                          // 32*9*4096 floats
    float* weff = s + (size_t)B_SZ * NG * HWSZ;          // 512*12 floats

    weff_kernel<<<(COUT * KPAD + 255) / 256, 256, 0, stream>>>(w_pw, weff);
    gsum_kernel<<<(B_SZ * NG * (HWSZ / 4)) / 256, 256, 0, stream>>>(x, s);
    shiftmm_kernel<<<B_SZ * HH, 256, 0, stream>>>(s, weff, out);
}